// GATv2Conv_24970939859605
// MI455X (gfx1250) — compile-verified
//
#include <hip/hip_runtime.h>
#include <hip/hip_bf16.h>

// ---------------------------------------------------------------------------
// GATv2Conv for MI455X (gfx1250, wave32).
//   N=50000, C_IN=128, H=4, D=32, E=800000 (+N self loops)
// Phase 1: xl = x@Wl^T, xr = x@Wr^T via v_wmma_f32_16x16x4_f32 (fp32 WMMA).
// Phase 2: per-edge score + segment softmax + scatter via global atomics.
// ---------------------------------------------------------------------------

typedef __attribute__((ext_vector_type(2))) float v2f;
typedef __attribute__((ext_vector_type(8))) float v8f;

#define HEADS 4
#define HDIM 32
#define CIN 128          // = HEADS*HDIM = input channels = xl/xr row width
#define NEG_SLOPE 0.2f

// ---------------------------------------------------------------------------
// zero fill (grid stride)
// ---------------------------------------------------------------------------
__global__ void gat_zero_kernel(float* __restrict__ p, size_t n) {
    size_t i = (size_t)blockIdx.x * blockDim.x + threadIdx.x;
    size_t stride = (size_t)gridDim.x * blockDim.x;
    for (; i < n; i += stride) p[i] = 0.0f;
}

// ---------------------------------------------------------------------------
// xl = X @ Wl^T, xr = X @ Wr^T.  X: Nn x 128 row-major, W*: 128 x 128 row-major.
// One wave computes a 16x16 output tile, K=128 in 32 steps of v_wmma 16x16x4.
// blockDim = 256 (8 waves); wave w owns column tile w (8 tiles of 16 = 128 cols).
// blockIdx.x = row tile, blockIdx.y selects (Wl->xl) vs (Wr->xr).
//
// fp32 WMMA fragment layout (CDNA5 ISA, 32-bit 16x4 A / 4x16 B / 16x16 C):
//   A: lane<16: m=lane, holds K=k0+0,k0+1 ; lane>=16: m=lane-16, K=k0+2,k0+3
//   B: lane<16: n=lane, holds K=k0+0,k0+1 ; lane>=16: n=lane-16, K=k0+2,k0+3
//      (B[k][n] = W[n][k] since out = X @ W^T)
//   C: vgpr i, lane<16 -> row i,  col lane ; lane>=16 -> row i+8, col lane-16
// ---------------------------------------------------------------------------
__global__ __launch_bounds__(256) void gat_gemm_wmma(
    const float* __restrict__ X,
    const float* __restrict__ Wl,
    const float* __restrict__ Wr,
    float* __restrict__ XL,
    float* __restrict__ XR,
    int Nn)
{
    const int wave = threadIdx.x >> 5;           // 0..7 : column tile
    const int lane = threadIdx.x & 31;
    const int row0 = blockIdx.x * 16;
    const int col0 = wave * 16;

    const float* __restrict__ W  = (blockIdx.y == 0) ? Wl : Wr;
    float*       __restrict__ O  = (blockIdx.y == 0) ? XL : XR;

    const int m  = lane & 15;
    const int hi = lane >> 4;                    // 0 or 1 -> K offset 0 or 2

    int arow = row0 + m; if (arow >= Nn) arow = Nn - 1;   // clamp (N%16==0 normally)

    const float* __restrict__ ap = X + (size_t)arow * CIN;
    const float* __restrict__ bp = W + (size_t)(col0 + m) * CIN;

    v8f acc = {};
    #pragma unroll
    for (int k0 = 0; k0 < CIN; k0 += 4) {
        const int ka = k0 + hi * 2;
        v2f a, b;
        a.x = ap[ka];  a.y = ap[ka + 1];
        b.x = bp[ka];  b.y = bp[ka + 1];
        // D = A x B + C ; emits v_wmma_f32_16x16x4_f32
        acc = __builtin_amdgcn_wmma_f32_16x16x4_f32(
            /*neg_a=*/false, a, /*neg_b=*/false, b,
            /*c_mod=*/(short)0, acc, /*reuse_a=*/false, /*reuse_b=*/false);
    }

    #pragma unroll
    for (int i = 0; i < 8; ++i) {
        const int r = row0 + i + hi * 8;
        if (r < Nn) O[(size_t)r * CIN + col0 + m] = acc[i];
    }
}

// ---------------------------------------------------------------------------
// Per-edge raw score e[eid][h] = sum_d att[h][d]*leakyrelu(xl[s][h][d]+xr[d][h][d])
// plus running segment max (against 0) via int-bitpattern atomicMax
// (valid because scores <= 0 never beat the 0.0f init, and positive IEEE
//  floats order-preserve as ints).
// One thread per (edge, head). Edge id >= E encodes self-loop (s=d=eid-E).
// ---------------------------------------------------------------------------
__global__ void gat_edge_score(
    const float* __restrict__ xl, const float* __restrict__ xr,
    const int* __restrict__ src, const int* __restrict__ dst,
    const float* __restrict__ att,
    float* __restrict__ ebuf, int* __restrict__ emax_i,
    int Etot, int E)
{
    const int t = blockIdx.x * blockDim.x + threadIdx.x;
    if (t >= Etot * HEADS) return;
    const int eid = t >> 2;
    const int h   = t & 3;

    int s, d;
    if (eid < E) { s = src[eid]; d = dst[eid]; }
    else         { s = d = eid - E; }

    const float* __restrict__ pl = xl + (size_t)s * CIN + h * HDIM;
    const float* __restrict__ pr = xr + (size_t)d * CIN + h * HDIM;
    const float* __restrict__ pa = att + h * HDIM;

    float acc = 0.0f;
    #pragma unroll
    for (int i = 0; i < HDIM; ++i) {
        float v = pl[i] + pr[i];
        v = (v > 0.0f) ? v : NEG_SLOPE * v;
        acc += pa[i] * v;
    }
    ebuf[(size_t)eid * HEADS + h] = acc;
    if (acc > 0.0f)
        atomicMax(&emax_i[d * HEADS + h], __float_as_int(acc));
}

// ---------------------------------------------------------------------------
// exp(e - emax[dst]) in place + atomic denom accumulation.
// emax buffer holds int bits of non-negative floats -> reads back correctly
// as float (0x00000000 == 0.0f).
// ---------------------------------------------------------------------------
__global__ void gat_edge_exp(
    const int* __restrict__ dst,
    float* __restrict__ ebuf,
    const float* __restrict__ emax,
    float* __restrict__ denom,
    int Etot, int E)
{
    const int t = blockIdx.x * blockDim.x + threadIdx.x;
    if (t >= Etot * HEADS) return;
    const int eid = t >> 2;
    const int h   = t & 3;
    const int d   = (eid < E) ? dst[eid] : (eid - E);

    const float m  = emax[d * HEADS + h];
    const float ex = __expf(ebuf[(size_t)eid * HEADS + h] - m);
    ebuf[(size_t)eid * HEADS + h] = ex;
    atomicAdd(&denom[d * HEADS + h], ex);
}

// ---------------------------------------------------------------------------
// alpha-weighted scatter: accum[dst][h][:] += xr[src][h][:] * alpha
// One thread per (edge, head), 32 f32 atomics each.
// ---------------------------------------------------------------------------
__global__ void gat_edge_scatter(
    const float* __restrict__ xr,
    const int* __restrict__ src, const int* __restrict__ dst,
    const float* __restrict__ ebuf, const float* __restrict__ denom,
    float* __restrict__ accum,
    int Etot, int E)
{
    const int t = blockIdx.x * blockDim.x + threadIdx.x;
    if (t >= Etot * HEADS) return;
    const int eid = t >> 2;
    const int h   = t & 3;

    int s, d;
    if (eid < E) { s = src[eid]; d = dst[eid]; }
    else         { s = d = eid - E; }

    float den = denom[d * HEADS + h];
    den = (den > 1e-16f) ? den : 1e-16f;
    const float alpha = ebuf[(size_t)eid * HEADS + h] / den;

    const float* __restrict__ p = xr    + (size_t)s * CIN + h * HDIM;
    float*       __restrict__ o = accum + (size_t)d * CIN + h * HDIM;
    #pragma unroll
    for (int i = 0; i < HDIM; ++i)
        atomicAdd(&o[i], p[i] * alpha);
}

// ---------------------------------------------------------------------------
// out[n][d] = mean_h accum[n][h][d] + bias[d]
// ---------------------------------------------------------------------------
__global__ void gat_finalize(
    const float* __restrict__ accum, const float* __restrict__ bias,
    float* __restrict__ out, int Nn)
{
    const int t = blockIdx.x * blockDim.x + threadIdx.x;
    if (t >= Nn * HDIM) return;
    const int n  = t >> 5;
    const int dc = t & 31;
    const float* __restrict__ a = accum + (size_t)n * CIN + dc;
    out[t] = (a[0] + a[HDIM] + a[2 * HDIM] + a[3 * HDIM]) * 0.25f + bias[dc];
}

// ---------------------------------------------------------------------------
extern "C" void kernel_launch(void* const* d_in, const int* in_sizes, int n_in,
                              void* d_out, int out_size, void* d_ws, size_t ws_size,
                              hipStream_t stream)
{
    const float* x    = (const float*)d_in[0];   // (N, 128)
    const int*   eidx = (const int*)d_in[1];     // (2, E)
    const float* Wl   = (const float*)d_in[2];   // (128, 128)
    const float* Wr   = (const float*)d_in[3];   // (128, 128)
    const float* att  = (const float*)d_in[4];   // (1, 4, 32)
    const float* bias = (const float*)d_in[5];   // (32,)
    float* out = (float*)d_out;                  // (N, 32)

    const int Nn   = in_sizes[0] / CIN;          // 50000
    const int E    = in_sizes[1] / 2;            // 800000
    const int Etot = E + Nn;                     // with self loops

    const int* src = eidx;
    const int* dst = eidx + E;

    // workspace layout (floats)
    float* ws = (float*)d_ws;
    size_t off = 0;
    float* xl    = ws + off; off += (size_t)Nn * CIN;     // N x 128
    float* xr    = ws + off; off += (size_t)Nn * CIN;     // N x 128
    float* ebuf  = ws + off; off += (size_t)Etot * HEADS; // per-edge scores / exp
    float* emax  = ws + off; off += (size_t)Nn * HEADS;   // segment max (int bits)
    float* denom = ws + off; off += (size_t)Nn * HEADS;   // segment sum
    float* accum = ws + off; off += (size_t)Nn * CIN;     // N x H x D scatter target
    (void)ws_size;

    // 1) zero emax + denom + accum (contiguous region starting at emax)
    {
        size_t nz = (size_t)Nn * HEADS * 2 + (size_t)Nn * CIN;
        int blocks = 2048;
        gat_zero_kernel<<<blocks, 256, 0, stream>>>(emax, nz);
    }

    // 2) WMMA GEMMs: xl = x@Wl^T, xr = x@Wr^T
    {
        dim3 grid((Nn + 15) / 16, 2);
        gat_gemm_wmma<<<grid, 256, 0, stream>>>(x, Wl, Wr, xl, xr, Nn);
    }

    const int edgeThreads = Etot * HEADS;
    const int edgeBlocks  = (edgeThreads + 255) / 256;

    // 3) raw scores + segment max
    gat_edge_score<<<edgeBlocks, 256, 0, stream>>>(
        xl, xr, src, dst, att, ebuf, (int*)emax, Etot, E);

    // 4) exp + denom
    gat_edge_exp<<<edgeBlocks, 256, 0, stream>>>(
        dst, ebuf, emax, denom, Etot, E);

    // 5) alpha-weighted scatter
    gat_edge_scatter<<<edgeBlocks, 256, 0, stream>>>(
        xr, src, dst, ebuf, denom, accum, Etot, E);

    // 6) head mean + bias
    {
        const int t = Nn * HDIM;
        gat_finalize<<<(t + 255) / 256, 256, 0, stream>>>(accum, bias, out, Nn);
    }
}